// GCN_scalar_86157043958238
// MI455X (gfx1250) — compile-verified
//
#include <hip/hip_runtime.h>
#include <hip/hip_bf16.h>

#define N_NODES 100000
#define N_FEAT  128
#define N_EDGES 1600000

typedef __attribute__((ext_vector_type(2))) float v2f;
typedef __attribute__((ext_vector_type(8))) float v8f;

// ---------------------------------------------------------------------------
// Zero-fill (float4 granularity, grid-stride)
// ---------------------------------------------------------------------------
__global__ void zero_kernel(float4* __restrict__ p, long long n4) {
    long long i = (long long)blockIdx.x * blockDim.x + threadIdx.x;
    long long stride = (long long)gridDim.x * blockDim.x;
    float4 z = make_float4(0.f, 0.f, 0.f, 0.f);
    for (; i < n4; i += stride) p[i] = z;
}

// ---------------------------------------------------------------------------
// Scatter-add aggregation: one wave per edge, float4 per lane (128 floats).
// feat rows and agg rows live in L2 (51.2 MB each << 192 MB), so the gather
// and the f32 atomics run at L2 bandwidth / atomic throughput.
// ---------------------------------------------------------------------------
__global__ void scatter_add_kernel(const float* __restrict__ feat,
                                   const int* __restrict__ edges,
                                   float* __restrict__ agg) {
    int wave = blockIdx.x * (blockDim.x >> 5) + (threadIdx.x >> 5);
    int lane = threadIdx.x & 31;
    if (wave >= N_EDGES) return;
    int s = edges[wave];
    int d = edges[N_EDGES + wave];
    const float4 v = ((const float4*)(feat + (size_t)s * N_FEAT))[lane];
    float* dp = agg + (size_t)d * N_FEAT + lane * 4;
    atomicAdd(dp + 0, v.x);
    atomicAdd(dp + 1, v.y);
    atomicAdd(dp + 2, v.z);
    atomicAdd(dp + 3, v.w);
}

// ---------------------------------------------------------------------------
// Fused dual-GEMM + bias + ReLU:
//   out = relu([agg | x] @ [W_rel | W_root]^T + b)
// FP32 WMMA (V_WMMA_F32_16X16X4_F32), one wave per 16-row panel computing
// the full 16x128 output stripe with 8 accumulators (one per 16-col tile).
// Per K-step: 1 A load + 8 B loads (instruction-offset addressed), then
// 8 independent WMMAs -> each load-wait amortized over 8 matrix ops, and
// the accumulation chains are independent (back-to-back XDL issue).
// Out-of-range rows are clamped to row 0: their garbage lands only in
// C rows that the row-guarded store never writes.
// A layout: lane(0..15)=row M, half-wave selects K pair {k,k+1}/{k+2,k+3}.
// B layout: lane%16 = col N, same half-wave K-pair mapping.
// C layout: VGPR r -> row (r + 8*half), col = lane%16.
// ---------------------------------------------------------------------------
__global__ __launch_bounds__(256)
void graphconv_gemm_kernel(const float* __restrict__ agg,
                           const float* __restrict__ xin,
                           const float* __restrict__ Wrel,
                           const float* __restrict__ Wroot,
                           const float* __restrict__ bias,
                           float* __restrict__ out) {
    const int lane = threadIdx.x & 31;
    const int wid  = threadIdx.x >> 5;
    const int half = lane >> 4;   // 0 or 1: selects K sub-pair
    const int l    = lane & 15;   // row index (A) / col index (B,C)
    const int m_base = (blockIdx.x * 8 + wid) * 16;
    const int arow_i = m_base + l;
    const int crow   = (arow_i < N_NODES) ? arow_i : 0;   // clamp, no predication
    const float* arow = agg  + (size_t)crow * N_FEAT + 2 * half;
    const float* xrow = xin  + (size_t)crow * N_FEAT + 2 * half;
    const float* wrb  = Wrel  + (size_t)l * N_FEAT + 2 * half;  // + nt*16*128 via imm offset
    const float* wob  = Wroot + (size_t)l * N_FEAT + 2 * half;

    v8f c[8] = {};

    // K-half 1: agg @ W_rel^T
    #pragma unroll
    for (int p = 0; p < 32; ++p) {
        float2 a = *(const float2*)(arow + 4 * p);
        float2 b[8];
        #pragma unroll
        for (int nt = 0; nt < 8; ++nt)
            b[nt] = *(const float2*)(wrb + nt * 16 * N_FEAT + 4 * p);
        #pragma unroll
        for (int nt = 0; nt < 8; ++nt) {
            v2f A = {a.x, a.y};
            v2f B = {b[nt].x, b[nt].y};
            c[nt] = __builtin_amdgcn_wmma_f32_16x16x4_f32(false, A, false, B,
                                                          (short)0, c[nt], false, false);
        }
    }
    // K-half 2: x @ W_root^T
    #pragma unroll
    for (int p = 0; p < 32; ++p) {
        float2 a = *(const float2*)(xrow + 4 * p);
        float2 b[8];
        #pragma unroll
        for (int nt = 0; nt < 8; ++nt)
            b[nt] = *(const float2*)(wob + nt * 16 * N_FEAT + 4 * p);
        #pragma unroll
        for (int nt = 0; nt < 8; ++nt) {
            v2f A = {a.x, a.y};
            v2f B = {b[nt].x, b[nt].y};
            c[nt] = __builtin_amdgcn_wmma_f32_16x16x4_f32(false, A, false, B,
                                                          (short)0, c[nt], false, false);
        }
    }

    // Bias + ReLU + store (row-guarded; half-waves cover rows r and r+8)
    #pragma unroll
    for (int nt = 0; nt < 8; ++nt) {
        const int col = nt * 16 + l;
        const float bcol = bias[col];
        #pragma unroll
        for (int r = 0; r < 8; ++r) {
            const int orow = m_base + r + 8 * half;
            if (orow < N_NODES) {
                out[(size_t)orow * N_FEAT + col] = fmaxf(c[nt][r] + bcol, 0.0f);
            }
        }
    }
}

// ---------------------------------------------------------------------------
// Out-degree histogram (for the fused layer-3 trick).
// ---------------------------------------------------------------------------
__global__ void degree_kernel(const int* __restrict__ edges,
                              int* __restrict__ deg) {
    int e = blockIdx.x * blockDim.x + threadIdx.x;
    if (e < N_EDGES) atomicAdd(&deg[edges[e]], 1);
}

// ---------------------------------------------------------------------------
// Column sums of h2: S[0..127] = sum_i h2[i,:], S[128..255] = sum_i deg[i]*h2[i,:]
// ---------------------------------------------------------------------------
#define CS_ROWS 256
__global__ void colsum_kernel(const float* __restrict__ h2,
                              const int* __restrict__ deg,
                              float* __restrict__ S) {
    const int f = threadIdx.x;  // 0..127
    int r0 = blockIdx.x * CS_ROWS;
    int r1 = r0 + CS_ROWS; if (r1 > N_NODES) r1 = N_NODES;
    float sroot = 0.f, srel = 0.f;
    for (int r = r0; r < r1; ++r) {
        float v = h2[(size_t)r * N_FEAT + f];
        sroot += v;
        srel  += (float)deg[r] * v;
    }
    atomicAdd(&S[f],       sroot);
    atomicAdd(&S[128 + f], srel);
}

// ---------------------------------------------------------------------------
// Final: out = (W3_rel . S_rel + W3_root . S_root) / N + b3
// ---------------------------------------------------------------------------
__global__ void final_kernel(const float* __restrict__ S,
                             const float* __restrict__ W3rel,
                             const float* __restrict__ W3root,
                             const float* __restrict__ b3,
                             float* __restrict__ out) {
    __shared__ float red[128];
    const int f = threadIdx.x;
    red[f] = W3rel[f] * S[128 + f] + W3root[f] * S[f];
    __syncthreads();
    for (int s = 64; s > 0; s >>= 1) {
        if (f < s) red[f] += red[f + s];
        __syncthreads();
    }
    if (f == 0) out[0] = red[0] / (float)N_NODES + b3[0];
}

// ---------------------------------------------------------------------------
extern "C" void kernel_launch(void* const* d_in, const int* in_sizes, int n_in,
                              void* d_out, int out_size, void* d_ws, size_t ws_size,
                              hipStream_t stream) {
    const float* x      = (const float*)d_in[0];
    const int*   edges  = (const int*)d_in[1];      // [2, N_EDGES], row0=src, row1=dst
    const float* W1rel  = (const float*)d_in[2];
    const float* b1     = (const float*)d_in[3];
    const float* W1root = (const float*)d_in[4];
    const float* W2rel  = (const float*)d_in[5];
    const float* b2     = (const float*)d_in[6];
    const float* W2root = (const float*)d_in[7];
    const float* W3rel  = (const float*)d_in[8];
    const float* b3     = (const float*)d_in[9];
    const float* W3root = (const float*)d_in[10];

    const size_t FEATB = (size_t)N_NODES * N_FEAT * sizeof(float);  // 51,200,000
    char* ws  = (char*)d_ws;
    float* agg = (float*)(ws);
    float* h1  = (float*)(ws + FEATB);
    float* h2  = (float*)(ws + 2 * FEATB);
    int*   deg = (int*)  (ws + 3 * FEATB);
    float* S   = (float*)(ws + 3 * FEATB + (size_t)N_NODES * sizeof(int));

    const long long feat_f4 = (long long)N_NODES * N_FEAT / 4;
    const int scatter_blocks = N_EDGES / 8;                // 8 waves/block, 1 edge/wave
    const int gemm_blocks    = (N_NODES + 127) / 128;      // 8 waves/block, 16 rows/wave

    // ---- Layer 1 ----
    zero_kernel<<<1024, 256, 0, stream>>>((float4*)agg, feat_f4);
    scatter_add_kernel<<<scatter_blocks, 256, 0, stream>>>(x, edges, agg);
    graphconv_gemm_kernel<<<gemm_blocks, 256, 0, stream>>>(agg, x, W1rel, W1root, b1, h1);

    // ---- Layer 2 ----
    zero_kernel<<<1024, 256, 0, stream>>>((float4*)agg, feat_f4);
    scatter_add_kernel<<<scatter_blocks, 256, 0, stream>>>(h1, edges, agg);
    graphconv_gemm_kernel<<<gemm_blocks, 256, 0, stream>>>(agg, h1, W2rel, W2root, b2, h2);

    // ---- Layer 3 fused with mean pool (linear layer => algebraic fusion) ----
    // zero deg (400,000 B) + S (1,024 B) in one float4 pass
    zero_kernel<<<64, 256, 0, stream>>>((float4*)deg,
        (long long)((N_NODES * sizeof(int) + 256 * sizeof(float)) / sizeof(float4)));
    degree_kernel<<<(N_EDGES + 255) / 256, 256, 0, stream>>>(edges, deg);
    colsum_kernel<<<(N_NODES + CS_ROWS - 1) / CS_ROWS, 128, 0, stream>>>(h2, deg, S);
    final_kernel<<<1, 128, 0, stream>>>(S, W3rel, W3root, b3, (float*)d_out);
}